// SinusoidalOscillator_10385230922332
// MI455X (gfx1250) — compile-verified
//
#include <hip/hip_runtime.h>
#include <hip/hip_bf16.h>
#include <cstddef>

#ifndef __has_builtin
#define __has_builtin(x) 0
#endif

#define THREADS     256
#define WAVES       (THREADS / 32)
#define CHUNK       8192                 // floats per chunk (32 KB)
#define PER_THREAD  (CHUNK / THREADS)    // 32 floats / thread
#define L16         (PER_THREAD / 4)     // 8 x b128 async loads / thread / chunk

// ---------------- CDNA5 async global->LDS copy (16 bytes per lane) ----------------
typedef int v4i __attribute__((ext_vector_type(4)));

#if __has_builtin(__builtin_amdgcn_global_load_async_to_lds_b128)
typedef __attribute__((address_space(1))) v4i gv4i;
typedef __attribute__((address_space(3))) v4i lv4i;
__device__ __forceinline__ void async_copy_b128(const void* g, void* l) {
  __builtin_amdgcn_global_load_async_to_lds_b128((gv4i*)g, (lv4i*)l, 0, 0);
}
#else
__device__ __forceinline__ void async_copy_b128(const void* g, void* l) {
  unsigned lds_off = (unsigned)(size_t)l;           // low 32 bits of flat LDS addr = LDS byte offset
  unsigned long long ga = (unsigned long long)(size_t)g;
  asm volatile("global_load_async_to_lds_b128 %0, %1, off"
               :: "v"(lds_off), "v"(ga) : "memory");
}
#endif

#if __has_builtin(__builtin_amdgcn_s_wait_asynccnt)
#define WAIT_ASYNC(n) __builtin_amdgcn_s_wait_asynccnt(n)
#else
#define WAIT_ASYNC(n) asm volatile("s_wait_asynccnt %0" :: "n"(n) : "memory")
#endif

// WAR fence: LDS reads must have completed before the async engine may
// overwrite the same buffer (async LDS writes are unordered vs DS ops).
#if __has_builtin(__builtin_amdgcn_s_wait_dscnt)
#define WAIT_DS0() __builtin_amdgcn_s_wait_dscnt(0)
#else
#define WAIT_DS0() asm volatile("s_wait_dscnt 0x0" ::: "memory")
#endif

// ---------------- helpers ----------------
__device__ __forceinline__ float fast_sin(float ph) {
  // reduce to [-pi, pi) then hardware v_sin_f32
  float t = ph * 0.15915494309189535f;   // /(2*pi)
  t = t - rintf(t);
  return __sinf(t * 6.283185307179586f);
}

__device__ __forceinline__ void issue_chunk(const float* rowp, float* lbuf,
                                            int elem_off, int tid) {
  const float* g = rowp + elem_off + tid * PER_THREAD;
  float*       l = lbuf + tid * PER_THREAD;
#pragma unroll
  for (int j = 0; j < L16; ++j) {
    async_copy_b128(g + 4 * j, l + 4 * j);
  }
}

// ---------------- main kernel: one workgroup per row ----------------
__global__ __launch_bounds__(THREADS) void osc_scan_kernel(
    const float* __restrict__ freq, const float* __restrict__ mod_env,
    const float* __restrict__ mod_amount, float* __restrict__ out, int N) {
  __shared__ float buf[2][CHUNK];
  __shared__ float wsum[WAVES];

  const int row  = blockIdx.x;
  const int tid  = threadIdx.x;
  const int lane = tid & 31;
  const int wave = tid >> 5;

  const float f0    = (6.283185307179586f / 48000.0f) * (20.0f + 1980.0f * freq[row]);
  const float f0amt = f0 * (-1.0f + 3.0f * mod_amount[row]);

  const float* rowp = mod_env + (size_t)row * (size_t)N;
  float*       outp = out     + (size_t)row * (size_t)N;
  const int nchunks = N / CHUNK;

  // prologue: prime the double buffer
  issue_chunk(rowp, &buf[0][0], 0, tid);
  if (nchunks > 1) issue_chunk(rowp, &buf[1][0], CHUNK, tid);

  float carry = 0.0f;
  for (int c = 0; c < nchunks; ++c) {
    float* lbuf = &buf[c & 1][0];

    // Async loads complete in order: <=8 outstanding ==> chunk c fully in LDS.
    if (c + 1 < nchunks) { WAIT_ASYNC(L16); } else { WAIT_ASYNC(0); }

    // pass 1: pull this thread's 32 values into registers, local sum
    float vals[PER_THREAD];
    float s = 0.0f;
    const float4* lv = (const float4*)(lbuf + tid * PER_THREAD);
#pragma unroll
    for (int j = 0; j < L16; ++j) {
      float4 v = lv[j];
      vals[4 * j + 0] = v.x; vals[4 * j + 1] = v.y;
      vals[4 * j + 2] = v.z; vals[4 * j + 3] = v.w;
      s += (v.x + v.y) + (v.z + v.w);
    }

    // buffer is free for this thread after its DS reads retire ->
    // fence, then start streaming chunk c+2 into it
    if (c + 2 < nchunks) {
      WAIT_DS0();
      issue_chunk(rowp, lbuf, (c + 2) * CHUNK, tid);
    }

    // wave-level inclusive scan of per-thread sums (wave32 shuffles)
    float incl = s;
#pragma unroll
    for (int off = 1; off < 32; off <<= 1) {
      float t = __shfl_up(incl, (unsigned)off, 32);
      if (lane >= off) incl += t;
    }
    float excl = incl - s;
    if (lane == 31) wsum[wave] = incl;
    __syncthreads();

    float woff = 0.0f, ctot = 0.0f;
#pragma unroll
    for (int w = 0; w < WAVES; ++w) {
      float t = wsum[w];
      ctot += t;
      if (w < wave) woff += t;
    }
    __syncthreads();  // wsum reusable next chunk

    // pass 2: exclusive running scan -> phase -> sin, B128 stores
    float running   = carry + woff + excl;
    const int gbase = c * CHUNK + tid * PER_THREAD;
    float4* ov      = (float4*)(outp + gbase);
#pragma unroll
    for (int j4 = 0; j4 < PER_THREAD / 4; ++j4) {
      float4 o;
      o.x = fast_sin(f0 * (float)(gbase + 4 * j4 + 0) + f0amt * running); running += vals[4 * j4 + 0];
      o.y = fast_sin(f0 * (float)(gbase + 4 * j4 + 1) + f0amt * running); running += vals[4 * j4 + 1];
      o.z = fast_sin(f0 * (float)(gbase + 4 * j4 + 2) + f0amt * running); running += vals[4 * j4 + 2];
      o.w = fast_sin(f0 * (float)(gbase + 4 * j4 + 3) + f0amt * running); running += vals[4 * j4 + 3];
      ov[j4] = o;
    }
    carry += ctot;
  }
}

// correctness fallback for N not divisible by CHUNK (not expected: N=65536)
__global__ void osc_naive_kernel(const float* __restrict__ freq,
                                 const float* __restrict__ mod_env,
                                 const float* __restrict__ mod_amount,
                                 float* __restrict__ out, int N) {
  const int row = blockIdx.x;
  if (threadIdx.x != 0) return;
  const float f0    = (6.283185307179586f / 48000.0f) * (20.0f + 1980.0f * freq[row]);
  const float f0amt = f0 * (-1.0f + 3.0f * mod_amount[row]);
  const float* rowp = mod_env + (size_t)row * (size_t)N;
  float*       outp = out     + (size_t)row * (size_t)N;
  float run = 0.0f;
  for (int n = 0; n < N; ++n) {
    outp[n] = fast_sin(f0 * (float)n + f0amt * run);
    run += rowp[n];
  }
}

extern "C" void kernel_launch(void* const* d_in, const int* in_sizes, int n_in,
                              void* d_out, int out_size, void* d_ws, size_t ws_size,
                              hipStream_t stream) {
  // setup_inputs order: [0] num_samples (int scalar), [1] freq [B], [2] mod_env [B*N], [3] mod_amount [B]
  const float* freq       = (const float*)d_in[1];
  const float* mod_env    = (const float*)d_in[2];
  const float* mod_amount = (const float*)d_in[3];
  float* out = (float*)d_out;

  const int B = in_sizes[1];
  const int N = in_sizes[2] / B;

  if ((N % CHUNK) == 0) {
    osc_scan_kernel<<<B, THREADS, 0, stream>>>(freq, mod_env, mod_amount, out, N);
  } else {
    osc_naive_kernel<<<B, 32, 0, stream>>>(freq, mod_env, mod_amount, out, N);
  }
}